// DGNLayerSimple_27582279975065
// MI455X (gfx1250) — compile-verified
//
#include <hip/hip_runtime.h>
#include <math.h>

#define N_NODES 50000
#define N_EDGES 1600000
#define D 128
#define AVG_D_LOG 3.4965075811f   /* log(33.0) */
#define BN_EPS 1e-5f

typedef __attribute__((ext_vector_type(2))) float v2f;
typedef __attribute__((ext_vector_type(4))) float v4f;
typedef __attribute__((ext_vector_type(8))) float v8f;

// Order-preserving float -> uint mapping so unsigned atomicMax == float max.
__device__ __forceinline__ unsigned enc_f32(float f) {
  unsigned u = __float_as_uint(f);
  return (u & 0x80000000u) ? ~u : (u | 0x80000000u);
}
__device__ __forceinline__ float dec_f32(unsigned k) {
  unsigned u = (k & 0x80000000u) ? (k & 0x7fffffffu) : ~k;
  return __uint_as_float(u);
}

// ---------------- Kernel 1: init accumulators ----------------
__global__ void k_init(float* __restrict__ sum, unsigned* __restrict__ maxk,
                       float* __restrict__ dir, float* __restrict__ deg,
                       float* __restrict__ den) {
  int i = blockIdx.x * blockDim.x + threadIdx.x;
  if (i < N_NODES * D) {
    sum[i] = 0.0f;
    dir[i] = 0.0f;
    maxk[i] = 0x007FFFFFu;  // enc(-inf)
  }
  if (i < N_NODES) {
    deg[i] = 0.0f;
    den[i] = 0.0f;
  }
}

// ---------------- Kernel 2: W -> WMMA-fragment-major swizzle ----------------
// Wf layout: [(ct*3+set)*48 + k2][lane*4 + j], 128 floats per row (one wave-wide
// b128 load = fragments for TWO consecutive K-steps of V_WMMA_F32_16X16X4_F32).
// Source element: row = set*384 + (k2*2 + (j>>1))*4 + 2*(lane>>4) + (j&1),
//                 col = ct*16 + (lane&15).
__global__ void k_wswz(const float* __restrict__ W, float* __restrict__ Wf) {
  int idx = blockIdx.x * blockDim.x + threadIdx.x;
  if (idx >= 8 * 3 * 48 * 128) return;
  int j    = idx & 3;
  int lane = (idx >> 2) & 31;
  int t    = idx >> 7;        // (ct*3+set)*48 + k2
  int k2   = t % 48;
  int cs   = t / 48;
  int set  = cs % 3;
  int ct   = cs / 3;
  int half = lane >> 4;
  int m    = lane & 15;
  int row  = set * 384 + (k2 * 2 + (j >> 1)) * 4 + 2 * half + (j & 1);
  int col  = ct * 16 + m;
  Wf[idx] = W[(size_t)row * D + col];
}

// ---------------- Kernel 3: edge scatter (L2-resident atomics) ----------------
__global__ void k_edge(const float* __restrict__ h, const float* __restrict__ eig,
                       const int* __restrict__ esrc, const int* __restrict__ edst,
                       float* sum, unsigned* maxk, float* dir,
                       float* deg, float* den) {
  int e = blockIdx.x * blockDim.y + threadIdx.y;
  if (e >= N_EDGES) return;
  int t = threadIdx.x;               // feature 0..127
  int s = esrc[e];
  int d = edst[e];
  float w = fabsf(eig[s * 4] - eig[d * 4]);   // only eig column 0 is used
  float v = h[(size_t)s * D + t];             // coalesced 512B row, hits L2
  size_t o = (size_t)d * D + t;
  atomicAdd(&sum[o], v);
  atomicMax(&maxk[o], enc_f32(v));
  atomicAdd(&dir[o], v * w);
  if (t == 0) {
    atomicAdd(&deg[d], 1.0f);
    atomicAdd(&den[d], w);
  }
}

// ---------------- Kernel 4: node phase — fused scaler-GEMM via f32 WMMA ----------------
// One wave32 per block, 16 nodes per wave.
// out = agg@W1 + s_amp*(agg@W2) + s_att*(agg@W3), then bias/snorm/BN/ReLU/residual.
__global__ void __launch_bounds__(32) k_node(
    const float* __restrict__ sumb, const unsigned* __restrict__ maxk,
    const float* __restrict__ dirb, const float* __restrict__ deg,
    const float* __restrict__ den,
    const float* __restrict__ h, const float* __restrict__ snorm,
    const float* __restrict__ Wf, const float* __restrict__ bias,
    const float* __restrict__ gam, const float* __restrict__ bet,
    const float* __restrict__ mu, const float* __restrict__ var,
    float* __restrict__ out) {
  const int STR = 388;                        // 384 + 4 pad: conflict-free A-frag reads
  __shared__ float aggs[16 * 388];            // 16 nodes x 384 agg features
  __shared__ float s_amp[16], s_att[16], s_sn[16];

  int nb = blockIdx.x * 16;
  int lane = threadIdx.x;                     // 0..31

  if (lane < 16) {
    float dg = deg[nb + lane];
    float logD = logf(dg + 1.0f);
    s_amp[lane] = logD * (1.0f / AVG_D_LOG);
    s_att[lane] = AVG_D_LOG / fmaxf(logD, 1e-6f);
    s_sn[lane] = snorm[nb + lane];
  }

  // Build agg rows: [mean | max(0 if empty) | dir_av]
  for (int idx = lane; idx < 16 * 384; idx += 32) {
    int r = idx / 384;
    int c = idx - r * 384;
    int n = nb + r;
    float dg = deg[n];
    float v;
    if (c < 128) {
      v = sumb[(size_t)n * D + c] / fmaxf(dg, 1.0f);
    } else if (c < 256) {
      float mv = dec_f32(maxk[(size_t)n * D + (c - 128)]);
      v = (dg > 0.0f) ? mv : 0.0f;
    } else {
      v = dirb[(size_t)n * D + (c - 256)] / (den[n] + 1e-30f);
    }
    aggs[r * STR + c] = v;
  }
  __syncthreads();

  int half = lane >> 4;   // K sub-pair selector for A/B fragments
  int m = lane & 15;      // row (A) / column (B,C,D) index

  for (int ct = 0; ct < 8; ++ct) {            // 8 column tiles of 16
    v8f c0 = {}; v8f c1 = {}; v8f c2 = {};
    int col = ct * 16 + m;
    const float* WfB = Wf + (size_t)(ct * 3) * 48 * 128 + lane * 4;
    for (int k2 = 0; k2 < 48; ++k2) {         // two K-steps per iteration
      int kc = k2 * 8 + 2 * half;
      // A frags (16x4 each) from LDS: contiguous b64 pairs, 16B apart.
      v2f a0; a0[0] = aggs[m * STR + kc];     a0[1] = aggs[m * STR + kc + 1];
      v2f a1; a1[0] = aggs[m * STR + kc + 4]; a1[1] = aggs[m * STR + kc + 5];
      // B frags: one coalesced b128 per set = two K-steps' 4x16 fragments.
      v4f b0 = *(const v4f*)(WfB + (size_t)(0 * 48 + k2) * 128);
      v4f b1 = *(const v4f*)(WfB + (size_t)(1 * 48 + k2) * 128);
      v4f b2 = *(const v4f*)(WfB + (size_t)(2 * 48 + k2) * 128);
      v2f b0l = __builtin_shufflevector(b0, b0, 0, 1);
      v2f b0h = __builtin_shufflevector(b0, b0, 2, 3);
      v2f b1l = __builtin_shufflevector(b1, b1, 0, 1);
      v2f b1h = __builtin_shufflevector(b1, b1, 2, 3);
      v2f b2l = __builtin_shufflevector(b2, b2, 0, 1);
      v2f b2h = __builtin_shufflevector(b2, b2, 2, 3);

      c0 = __builtin_amdgcn_wmma_f32_16x16x4_f32(false, a0, false, b0l, (short)0, c0, false, false);
      c1 = __builtin_amdgcn_wmma_f32_16x16x4_f32(false, a0, false, b1l, (short)0, c1, false, false);
      c2 = __builtin_amdgcn_wmma_f32_16x16x4_f32(false, a0, false, b2l, (short)0, c2, false, false);
      c0 = __builtin_amdgcn_wmma_f32_16x16x4_f32(false, a1, false, b0h, (short)0, c0, false, false);
      c1 = __builtin_amdgcn_wmma_f32_16x16x4_f32(false, a1, false, b1h, (short)0, c1, false, false);
      c2 = __builtin_amdgcn_wmma_f32_16x16x4_f32(false, a1, false, b2h, (short)0, c2, false, false);
    }

    // Fused epilogue straight from WMMA accumulators.
    float bc  = bias[col];
    float sc  = gam[col] * rsqrtf(var[col] + BN_EPS);
    float mc  = mu[col];
    float btc = bet[col];
#pragma unroll
    for (int i = 0; i < 8; ++i) {
      int mm = i + 8 * half;                  // C/D row: VGPR i, lane-half
      int node = nb + mm;
      float val = c0[i] + s_amp[mm] * c1[i] + s_att[mm] * c2[i] + bc;
      val = val * s_sn[mm];
      val = (val - mc) * sc + btc;
      val = fmaxf(val, 0.0f);
      out[(size_t)node * D + col] = h[(size_t)node * D + col] + val;
    }
  }
}

extern "C" void kernel_launch(void* const* d_in, const int* in_sizes, int n_in,
                              void* d_out, int out_size, void* d_ws, size_t ws_size,
                              hipStream_t stream) {
  const float* h     = (const float*)d_in[0];
  const float* eig   = (const float*)d_in[1];
  const float* snorm = (const float*)d_in[2];
  const float* W     = (const float*)d_in[3];
  const float* b     = (const float*)d_in[4];
  const float* gam   = (const float*)d_in[5];
  const float* bet   = (const float*)d_in[6];
  const float* mu    = (const float*)d_in[7];
  const float* var   = (const float*)d_in[8];
  const int*   esrc  = (const int*)d_in[9];
  const int*   edst  = (const int*)d_in[10];
  float* out = (float*)d_out;

  char* ws = (char*)d_ws;
  size_t off = 0;
  float*    sum  = (float*)(ws + off);    off += (size_t)N_NODES * D * sizeof(float);
  unsigned* maxk = (unsigned*)(ws + off); off += (size_t)N_NODES * D * sizeof(unsigned);
  float*    dir  = (float*)(ws + off);    off += (size_t)N_NODES * D * sizeof(float);
  float*    deg  = (float*)(ws + off);    off += (size_t)N_NODES * sizeof(float);
  float*    den  = (float*)(ws + off);    off += (size_t)N_NODES * sizeof(float);
  float*    Wf   = (float*)(ws + off);    off += (size_t)8 * 3 * 48 * 128 * sizeof(float);

  {
    int total = N_NODES * D;
    k_init<<<(total + 255) / 256, 256, 0, stream>>>(sum, maxk, dir, deg, den);
  }
  {
    int total = 8 * 3 * 48 * 128;  // 147456
    k_wswz<<<(total + 255) / 256, 256, 0, stream>>>(W, Wf);
  }
  {
    dim3 blk(128, 8);  // 128 feature lanes x 8 edges per block
    int nblk = (N_EDGES + 7) / 8;
    k_edge<<<nblk, blk, 0, stream>>>(h, eig, esrc, edst, sum, maxk, dir, deg, den);
  }
  {
    k_node<<<N_NODES / 16, 32, 0, stream>>>(sum, maxk, dir, deg, den,
                                            h, snorm, Wf, b, gam, bet, mu, var, out);
  }
}